// TransPoseNet_45500883534234
// MI455X (gfx1250) — compile-verified
//
#include <hip/hip_runtime.h>
#include <hip/hip_bf16.h>
#include <math.h>

// Problem dims (fixed by the reference)
#define BB 64
#define SS 32
#define DD 1024
#define HH 1024
#define HSTRIDE (HH + 4)   // pad so the two lane-halves store to disjoint LDS banks

typedef float v2f __attribute__((ext_vector_type(2)));
typedef float v8f __attribute__((ext_vector_type(8)));

__device__ __forceinline__ float gelu_exact(float x) {
    return 0.5f * x * (1.0f + erff(x * 0.70710678118654752f));
}

// ---------------------------------------------------------------------------
// Kernel 1: scene logits -> log_softmax (written to out) + argmax idx (to ws)
// One block per sample b. 256 threads = 8 waves; each wave does 4 scenes.
// ---------------------------------------------------------------------------
__global__ __launch_bounds__(256) void scene_logsoftmax_kernel(
    const float* __restrict__ ldt, const float* __restrict__ ldr,
    const float* __restrict__ sw,  const float* __restrict__ sb,
    float* __restrict__ out, int* __restrict__ idxbuf)
{
    __shared__ float logits[SS];
    const int b    = blockIdx.x;
    const int tid  = threadIdx.x;
    const int lane = tid & 31;
    const int wave = tid >> 5;

    #pragma unroll
    for (int q = 0; q < 4; ++q) {
        const int s = wave * 4 + q;
        const float* pt = ldt + ((size_t)b * SS + s) * DD;
        const float* pr = ldr + ((size_t)b * SS + s) * DD;
        float acc = 0.0f;
        for (int d = lane * 4; d < DD; d += 32 * 4) {
            float4 t  = *(const float4*)(pt + d);
            float4 r  = *(const float4*)(pr + d);
            float4 wt = *(const float4*)(sw + d);
            float4 wr = *(const float4*)(sw + DD + d);
            acc += t.x * wt.x + t.y * wt.y + t.z * wt.z + t.w * wt.w;
            acc += r.x * wr.x + r.y * wr.y + r.z * wr.z + r.w * wr.w;
        }
        #pragma unroll
        for (int off = 16; off; off >>= 1) acc += __shfl_xor(acc, off, 32);
        if (lane == 0) logits[s] = acc + sb[0];
    }
    __syncthreads();

    if (wave == 0) {
        float v = logits[lane];
        float m = v;
        #pragma unroll
        for (int off = 16; off; off >>= 1) m = fmaxf(m, __shfl_xor(m, off, 32));
        float e = expf(v - m);
        float ssum = e;
        #pragma unroll
        for (int off = 16; off; off >>= 1) ssum += __shfl_xor(ssum, off, 32);
        const float lse = logf(ssum);
        out[BB * 7 + b * SS + lane] = v - m - lse;   // scene_log_distr region

        float av = v; int ai = lane;                  // first-occurrence argmax
        #pragma unroll
        for (int off = 16; off; off >>= 1) {
            float ov = __shfl_xor(av, off, 32);
            int   oi = __shfl_xor(ai, off, 32);
            if (ov > av || (ov == av && oi < ai)) { av = ov; ai = oi; }
        }
        if (lane == 0) idxbuf[b] = ai;
    }
}

// ---------------------------------------------------------------------------
// Kernel 2: group samples by chosen scene into <=16-row tiles so each expert
// weight matrix is streamed from HBM exactly once (dedup: 512MB -> ~224MB).
// Serial (1 thread, 2048 iterations) => deterministic.
// ---------------------------------------------------------------------------
__global__ void compact_kernel(const int* __restrict__ idx,
                               int* __restrict__ ntiles,
                               int* __restrict__ tscene,
                               int* __restrict__ tcnt,
                               int* __restrict__ tsamp)
{
    if (threadIdx.x == 0 && blockIdx.x == 0) {
        int nt = 0;
        for (int s = 0; s < SS; ++s) {
            int cnt = 0, cur = -1;
            for (int b = 0; b < BB; ++b) {
                if (idx[b] == s) {
                    if ((cnt & 15) == 0) { cur = nt++; tscene[cur] = s; tcnt[cur] = 0; }
                    tsamp[cur * 16 + tcnt[cur]] = b;
                    tcnt[cur]++;
                    cnt++;
                }
            }
        }
        *ntiles = nt;
    }
}

// ---------------------------------------------------------------------------
// Kernel 3: expert pose heads as true 16x1024x1024 GEMM tiles via f32 WMMA.
// grid = (2 heads, 64 tile slots), 512 threads = 16 waves.
// A tile rows = up to 16 gathered descriptors (zero-padded), staged transposed
// in LDS:  gT[k][m]  (lane-halves read disjoint 16-bank groups -> no conflict).
// B = W1[s][k:k+4, j0:j0+16] streamed from global (coalesced 64B rows).
// D 16x16 tile layout (ISA 7.12.2): VGPR r, lanes 0-15 -> M=r, lanes 16-31 ->
// M=r+8; extract all rows, bias + exact GELU -> padded hbuf, then one wave per
// valid sample row does the tiny 1024->O second layer.
// ---------------------------------------------------------------------------
__global__ __launch_bounds__(512) void pose_head_kernel(
    const float* __restrict__ ldt, const float* __restrict__ ldr,
    const float* __restrict__ Wt1, const float* __restrict__ bt1,
    const float* __restrict__ Wt2, const float* __restrict__ bt2,
    const float* __restrict__ Wr1, const float* __restrict__ br1,
    const float* __restrict__ Wr2, const float* __restrict__ br2,
    const int* __restrict__ ntiles, const int* __restrict__ tscene,
    const int* __restrict__ tcnt,   const int* __restrict__ tsamp,
    float* __restrict__ out)
{
    extern __shared__ float smem[];
    float* gT   = smem;               // [DD][16]       (64 KB)
    float* hbuf = smem + 16 * DD;     // [16][HSTRIDE]  (64.25 KB)

    const int head = blockIdx.x;      // 0 = translation (O=3), 1 = rotation (O=4)
    const int tile = blockIdx.y;
    if (tile >= *ntiles) return;      // uniform exit for unused tile slots

    const int tid  = threadIdx.x;
    const int lane = tid & 31;
    const int wave = tid >> 5;

    const int s   = tscene[tile];
    const int cnt = tcnt[tile];

    const float* descs = head ? ldr : ldt;
    const float* W1 = (head ? Wr1 : Wt1) + (size_t)s * DD * HH;
    const float* b1 = (head ? br1 : bt1) + (size_t)s * HH;
    const int    O  = head ? 4 : 3;
    const float* W2 = (head ? Wr2 : Wt2) + (size_t)s * HH * O;
    const float* b2 = (head ? br2 : bt2) + (size_t)s * O;

    // ---- stage up to 16 descriptors, transposed: gT[d*16 + m] ----
    {
        const int m = wave;           // 16 waves <-> 16 A-tile rows
        if (m < cnt) {
            const int bm = tsamp[tile * 16 + m];
            const float* gp = descs + ((size_t)bm * SS + s) * DD;
            for (int d = lane; d < DD; d += 32) gT[d * 16 + m] = gp[d];
        } else {
            for (int d = lane; d < DD; d += 32) gT[d * 16 + m] = 0.0f;
        }
    }
    __syncthreads();

    // ---- layer 1 GEMM: each wave owns 4 column tiles j0 = (wave + 16*t)*16 ----
    const int half = lane >> 4;       // 0: K={k,k+1}   1: K={k+2,k+3}
    const int nl   = lane & 15;       // A row id (M) / B column within tile

    v8f acc[4] = {};
    for (int k = 0; k < DD; k += 4) {
        v2f a;
        a.x = gT[(k + 2 * half)     * 16 + nl];
        a.y = gT[(k + 2 * half + 1) * 16 + nl];
        const float* wrow = W1 + (size_t)(k + 2 * half) * HH;
        #pragma unroll
        for (int t = 0; t < 4; ++t) {
            const int j0 = (wave + 16 * t) * 16;
            v2f bm;
            bm.x = wrow[j0 + nl];            // row (k+2*half)
            bm.y = wrow[HH + j0 + nl];       // row (k+2*half+1)
            acc[t] = __builtin_amdgcn_wmma_f32_16x16x4_f32(
                false, a, false, bm, (short)0, acc[t], false, false);
        }
    }

    // ---- extract D tiles: bias + exact GELU -> hbuf[m][j] ----
    #pragma unroll
    for (int t = 0; t < 4; ++t) {
        const int j0 = (wave + 16 * t) * 16;
        const float bias = b1[j0 + nl];
        #pragma unroll
        for (int r = 0; r < 8; ++r) {
            const int m = r + 8 * half;      // D row held by this lane/VGPR
            hbuf[m * HSTRIDE + j0 + nl] = gelu_exact(acc[t][r] + bias);
        }
    }
    __syncthreads();

    // ---- layer 2: wave m handles sample row m (1024 -> O) ----
    if (wave < cnt) {
        const int bm = tsamp[tile * 16 + wave];
        float p[4] = {0.0f, 0.0f, 0.0f, 0.0f};
        for (int j = lane; j < HH; j += 32) {
            const float hv = hbuf[wave * HSTRIDE + j];
            #pragma unroll
            for (int o = 0; o < 4; ++o)
                if (o < O) p[o] += hv * W2[(size_t)j * O + o];
        }
        #pragma unroll
        for (int o = 0; o < 4; ++o) {
            #pragma unroll
            for (int off = 16; off; off >>= 1) p[o] += __shfl_xor(p[o], off, 32);
        }
        if (lane == 0) {
            const int base = bm * 7 + (head ? 3 : 0);
            #pragma unroll
            for (int o = 0; o < 4; ++o)
                if (o < O) out[base + o] = p[o] + b2[o];
        }
    }
}

// ---------------------------------------------------------------------------
extern "C" void kernel_launch(void* const* d_in, const int* in_sizes, int n_in,
                              void* d_out, int out_size, void* d_ws, size_t ws_size,
                              hipStream_t stream)
{
    const float* ldt = (const float*)d_in[0];
    const float* ldr = (const float*)d_in[1];
    const float* sw  = (const float*)d_in[2];
    const float* sb  = (const float*)d_in[3];
    const float* Wt1 = (const float*)d_in[4];
    const float* bt1 = (const float*)d_in[5];
    const float* Wt2 = (const float*)d_in[6];
    const float* bt2 = (const float*)d_in[7];
    const float* Wr1 = (const float*)d_in[8];
    const float* br1 = (const float*)d_in[9];
    const float* Wr2 = (const float*)d_in[10];
    const float* br2 = (const float*)d_in[11];

    float* out = (float*)d_out;

    // workspace layout (ints)
    int* ws      = (int*)d_ws;
    int* idxbuf  = ws + 0;      // [64]
    int* ntiles  = ws + 64;     // [1]
    int* tscene  = ws + 128;    // [64]
    int* tcnt    = ws + 192;    // [64]
    int* tsamp   = ws + 256;    // [64*16]

    scene_logsoftmax_kernel<<<BB, 256, 0, stream>>>(ldt, ldr, sw, sb, out, idxbuf);

    compact_kernel<<<1, 32, 0, stream>>>(idxbuf, ntiles, tscene, tcnt, tsamp);

    const size_t shmem = (size_t)(16 * DD + 16 * HSTRIDE) * sizeof(float);
    dim3 grid(2, BB);   // 64 tile slots covers the worst case (<=35 tiles)
    pose_head_kernel<<<grid, 512, shmem, stream>>>(ldt, ldr,
                                                   Wt1, bt1, Wt2, bt2,
                                                   Wr1, br1, Wr2, br2,
                                                   ntiles, tscene, tcnt, tsamp,
                                                   out);
}